// TBEInputPrepareReference_6038724018288
// MI455X (gfx1250) — compile-verified
//
#include <hip/hip_runtime.h>
#include <stdint.h>

#define BLK 256

// ---------------------------------------------------------------------------
// Kernel 1: bulk copy of 16B-aligned data via the CDNA5 async LDS DMA path.
// Each wave issues U batches of GLOBAL_LOAD_ASYNC_TO_LDS_B128 (no VGPR data),
// drains ASYNCcnt, then streams the LDS back out with
// GLOBAL_STORE_ASYNC_FROM_LDS_B128. Each wave only touches its own LDS slice,
// so no barriers are needed; the s_wait_asynccnt 0 before the next iteration
// protects the LDS buffer from being overwritten while stores are in flight.
// ---------------------------------------------------------------------------
__global__ void copy_async_b128(const uint32_t* __restrict__ src,
                                uint32_t* __restrict__ dst,
                                long n16 /* # of 16-byte chunks */) {
    constexpr int U = 4;
    __shared__ uint32_t smem[BLK * U * 4];  // 16 KB per block
    const unsigned lds_base =
        (unsigned)(uintptr_t)(&smem[0]);    // flat addr low 32 bits == LDS offset
    const int tid = threadIdx.x;
    long base = (long)blockIdx.x * (BLK * U);
    const long stride = (long)gridDim.x * (BLK * U);

    for (; base < n16; base += stride) {
#pragma unroll
        for (int u = 0; u < U; ++u) {
            long c = base + (long)u * BLK + tid;
            if (c < n16) {
                unsigned lds = lds_base + (unsigned)(u * BLK + tid) * 16u;
                uint64_t g = (uint64_t)(uintptr_t)(src + c * 4);
                asm volatile("global_load_async_to_lds_b128 %0, %1, off"
                             :: "v"(lds), "v"(g) : "memory");
            }
        }
        // Drain: async loads must complete before this wave reads the LDS.
        asm volatile("s_wait_asynccnt 0x0" ::: "memory");
#pragma unroll
        for (int u = 0; u < U; ++u) {
            long c = base + (long)u * BLK + tid;
            if (c < n16) {
                unsigned lds = lds_base + (unsigned)(u * BLK + tid) * 16u;
                uint64_t g = (uint64_t)(uintptr_t)(dst + c * 4);
                asm volatile("global_store_async_from_lds_b128 %0, %1, off"
                             :: "v"(g), "v"(lds) : "memory");
            }
        }
        // Drain stores before the LDS buffer is reused by the next iteration.
        asm volatile("s_wait_asynccnt 0x0" ::: "memory");
    }
}

// ---------------------------------------------------------------------------
// Kernel 2: combined_offsets[t*B + j] = offsets[t*(B+1) + j] + t*L  (j < B),
// plus trailing element T*L. Tiny (~131K elements).
// ---------------------------------------------------------------------------
__global__ void offsets_kernel(const int32_t* __restrict__ offs,
                               int32_t* __restrict__ out,
                               int T, int B, int L) {
    int i = blockIdx.x * blockDim.x + threadIdx.x;
    int n = T * B;
    if (i < n) {
        int t = i / B;
        int j = i - t * B;
        out[i] = offs[t * (B + 1) + j] + t * L;
    }
    if (i == 0) out[n] = T * L;
}

// ---------------------------------------------------------------------------
// Kernel 3: dword-granularity streaming copy (handles the 4-byte-only aligned
// weights destination). Lane-contiguous addressing -> every load/store
// instruction moves a fully coalesced 128B per wave; 8 words per thread give
// 8 loads in flight per lane; non-temporal hints bypass L2 retention since
// each byte is touched exactly once.
// ---------------------------------------------------------------------------
__global__ void copy_words_nt(const uint32_t* __restrict__ src,
                              uint32_t* __restrict__ dst, long n) {
    constexpr int U = 8;
    long base = (long)blockIdx.x * (BLK * U) + threadIdx.x;
    uint32_t v[U];
#pragma unroll
    for (int k = 0; k < U; ++k) {
        long idx = base + (long)k * BLK;
        if (idx < n) v[k] = __builtin_nontemporal_load(src + idx);
    }
#pragma unroll
    for (int k = 0; k < U; ++k) {
        long idx = base + (long)k * BLK;
        if (idx < n) __builtin_nontemporal_store(v[k], dst + idx);
    }
}

extern "C" void kernel_launch(void* const* d_in, const int* in_sizes, int n_in,
                              void* d_out, int out_size, void* d_ws, size_t ws_size,
                              hipStream_t stream) {
    (void)n_in; (void)out_size; (void)d_ws; (void)ws_size;

    const uint32_t* indices = (const uint32_t*)d_in[0];   // int32 [T*L]
    const int32_t*  offsets = (const int32_t*)d_in[1];    // int32 [T*(B+1)]
    const uint32_t* weights = (const uint32_t*)d_in[2];   // fp32  [T*L]
    uint32_t* out = (uint32_t*)d_out;

    const int  T   = 8;                       // fixed by the reference problem
    const long TL  = (long)in_sizes[0];       // T*L
    const int  L   = (int)(TL / T);
    const int  Bp1 = in_sizes[1] / T;         // B+1
    const int  B   = Bp1 - 1;
    const long TB  = (long)T * B;

    // --- 1) combined_indices: async-DMA copy of 16B chunks + scalar tail ---
    long n16  = TL >> 2;                      // 16M words -> 4M chunks (exact here)
    long tail = TL - (n16 << 2);
    {
        long chunks_per_blk = (long)BLK * 4;
        int blocks = (int)((n16 + chunks_per_blk - 1) / chunks_per_blk);
        copy_async_b128<<<blocks, BLK, 0, stream>>>(indices, out, n16);
        if (tail) {
            copy_words_nt<<<1, BLK, 0, stream>>>(indices + (n16 << 2),
                                                 out + (n16 << 2), tail);
        }
    }

    // --- 2) combined_offsets: out[TL .. TL+TB] ---
    {
        int nOff = (int)TB;
        int blocks = (nOff + BLK - 1) / BLK;
        if (blocks < 1) blocks = 1;
        offsets_kernel<<<blocks, BLK, 0, stream>>>(offsets,
                                                   (int32_t*)(out + TL), T, B, L);
    }

    // --- 3) combined_weights: dst starts at word TL+TB+1 (only 4B aligned) ---
    {
        uint32_t* wdst = out + TL + TB + 1;
        long words_per_blk = (long)BLK * 8;
        int blocks = (int)((TL + words_per_blk - 1) / words_per_blk);
        copy_words_nt<<<blocks, BLK, 0, stream>>>(weights, wdst, TL);
    }
}